// Contrastive_Loss_16827681866035
// MI455X (gfx1250) — compile-verified
//
#include <hip/hip_runtime.h>

#define NW 16
#define NG 5
#define NF 10
#define STEP (1 + NG + NF)        /* 16 */
#define NPAIRS (NW * STEP)        /* 256 */
#define T_LEN 400
#define DIMK 64
#define NT (T_LEN / 16)           /* 25 tiles per side */
#define NTHREADS 512
#define NWAVES (NTHREADS / 32)    /* 16 waves */
#define GAMMA_F 5.0f
#define INVG_F 0.2f
#define BIG_F 1.0e10f

typedef __attribute__((ext_vector_type(2))) float v2f;
typedef __attribute__((ext_vector_type(8))) float v8f;

__device__ __forceinline__ float softmin3(float a, float b, float c) {
  // -gamma * log(exp(-a/g)+exp(-b/g)+exp(-c/g)), max-shifted for stability
  float m = fminf(a, fminf(b, c));
  float s = __expf((m - a) * INVG_F) + __expf((m - b) * INVG_F) +
            __expf((m - c) * INVG_F);
  return m - GAMMA_F * __logf(s);
}

// CDNA5 async direct-to-LDS copy (ASYNCcnt-tracked, bypasses VGPR data path).
__device__ __forceinline__ void async_copy_b128(unsigned lds_off,
                                                const void* gaddr) {
  asm volatile("global_load_async_to_lds_b128 %0, %1, off"
               :
               : "v"(lds_off), "v"((unsigned long long)(uintptr_t)gaddr)
               : "memory");
}
__device__ __forceinline__ void wait_asynccnt0() {
  asm volatile("s_wait_asynccnt 0" ::: "memory");
}

// One workgroup per pair: fused (A·B^T cost tile via fp32 WMMA) + blocked
// soft-DTW wavefront, everything resident in the 320KB WGP LDS.
__global__ __launch_bounds__(NTHREADS, 1)
void sdtw_pair_kernel(const float* __restrict__ data,
                      const int* __restrict__ lens,
                      float* __restrict__ pairDist) {
  __shared__ __align__(16) float sA[T_LEN * DIMK];   // 100 KB
  __shared__ __align__(16) float sB[T_LEN * DIMK];   // 100 KB
  __shared__ float sNA[T_LEN];
  __shared__ float sNB[T_LEN];
  __shared__ float sROWB[T_LEN + 1];   // bottom-row boundary per tile column
  __shared__ float sCOLB[T_LEN + 1];   // right-col boundary per tile row
  __shared__ float sCORN[3 * 32];      // tile corners, triple-buffered by diag%3
  __shared__ float sDt[NWAVES * 16 * 18]; // per-wave 16x16 D tile (pad 18)
  __shared__ float sRes;

  const int tid  = threadIdx.x;
  const int wave = tid >> 5;
  const int lane = tid & 31;
  const int r16  = lane & 15;   // row (A frag) / col (B frag) index
  const int kh   = lane >> 4;   // K half select per ISA f32 A/B layout

  const int p = blockIdx.x;
  const int g = p / STEP;
  const float* Aseq = data + (size_t)(g * STEP) * T_LEN * DIMK; // anchor seq
  const float* Bseq = data + (size_t)p * T_LEN * DIMK;
  const int la = lens[g * STEP];
  const int lb = lens[p];

  // ---- stage A,B into LDS via async B128 copies (no VGPR data path) ----
  {
    const float4* gA = reinterpret_cast<const float4*>(Aseq);
    const float4* gB = reinterpret_cast<const float4*>(Bseq);
    const float4* s4A = reinterpret_cast<const float4*>(sA);
    const float4* s4B = reinterpret_cast<const float4*>(sB);
    for (int i = tid; i < T_LEN * DIMK / 4; i += NTHREADS) {
      async_copy_b128((unsigned)(uintptr_t)(s4A + i), gA + i);
      async_copy_b128((unsigned)(uintptr_t)(s4B + i), gB + i);
    }
  }
  for (int i = tid; i <= T_LEN; i += NTHREADS) {
    sROWB[i] = BIG_F;   // R[0][j] = BIG
    sCOLB[i] = BIG_F;   // R[i][0] = BIG
  }
  if (tid == 0) sRes = 0.0f;
  wait_asynccnt0();     // this wave's async copies landed in LDS
  __syncthreads();      // publish to all waves

  // ---- row norms ----
  for (int r = tid; r < T_LEN; r += NTHREADS) {
    float sa = 0.f, sb = 0.f;
    const float* ar = &sA[r * DIMK];
    const float* br = &sB[r * DIMK];
#pragma unroll 8
    for (int k = 0; k < DIMK; ++k) {
      sa += ar[k] * ar[k];
      sb += br[k] * br[k];
    }
    sNA[r] = sa;
    sNB[r] = sb;
  }
  __syncthreads();

  float* dt = &sDt[wave * 16 * 18];

  // ---- blocked wavefront over 25x25 tiles ----
  for (int dd = 0; dd < 2 * NT - 1; ++dd) {
    int tjmin = (dd > NT - 1) ? dd - (NT - 1) : 0;
    int tjmax = (dd < NT - 1) ? dd : NT - 1;
    int cnt = tjmax - tjmin + 1;

    for (int t = wave; t < cnt; t += NWAVES) {
      int tj = tjmin + t;
      int ti = dd - tj;

      // ---- 16x16 D tile: C = A_tile(16x64) * B_tile^T(64x16), fp32 WMMA ----
      const float* aRow = &sA[(ti * 16 + r16) * DIMK];
      const float* bRow = &sB[(tj * 16 + r16) * DIMK];
      v8f acc = {};
#pragma unroll
      for (int k0 = 0; k0 < DIMK; k0 += 4) {
        int kk = k0 + 2 * kh;
        v2f af; af[0] = aRow[kk]; af[1] = aRow[kk + 1];
        v2f bf; bf[0] = bRow[kk]; bf[1] = bRow[kk + 1];
        acc = __builtin_amdgcn_wmma_f32_16x16x4_f32(
            false, af, false, bf, (short)0, acc, false, false);
      }
      // D = na + nb - 2*dot ; C layout: lane -> col, vgpr v -> row v + 8*kh
      float nbv = sNB[tj * 16 + r16];
#pragma unroll
      for (int v = 0; v < 8; ++v) {
        int rr = v + 8 * kh;
        dt[rr * 18 + r16] = sNA[ti * 16 + rr] + nbv - 2.0f * acc[v];
      }

      // ---- corner (written 2 diagonals ago -> buffer (dd+1)%3) ----
      float corner;
      if (ti == 0 && tj == 0) corner = 0.0f;
      else if (ti == 0 || tj == 0) corner = BIG_F;
      else corner = sCORN[((dd + 1) % 3) * 32 + tj];

      // ---- intra-tile DP: lane i owns row i, 31 shuffle-pipelined steps ----
      float prev = sCOLB[ti * 16 + 1 + r16];  // R[gi][tj*16] (left boundary)
      float oldp = prev;
#pragma unroll 1
      for (int s = 0; s < 31; ++s) {
        float up = __shfl_up(prev, 1);   // R[i-1][j]
        float dg = __shfl_up(oldp, 1);   // R[i-1][j-1]
        if (r16 == 0 && s < 16) {        // top boundary for row 0
          up = sROWB[tj * 16 + 1 + s];
          dg = (s == 0) ? corner : sROWB[tj * 16 + s];
        }
        int j = s - r16;
        if (lane < 16 && j >= 0 && j < 16) {
          float dv = dt[r16 * 18 + j];
          float nv = dv + softmin3(dg, up, prev);
          oldp = prev;
          prev = nv;
          if (r16 == 15) {               // bottom row -> next tile row's top
            sROWB[tj * 16 + 1 + j] = nv;
            if (j == 15) sCORN[(dd % 3) * 32 + tj + 1] = nv;
          }
          int gi = ti * 16 + 1 + r16;
          int gj = tj * 16 + 1 + j;
          if (gi == la && gj == lb) sRes = nv;  // R[la][lb]
        }
      }
      if (lane < 16) sCOLB[ti * 16 + 1 + r16] = prev;  // right column
    }
    __syncthreads();
  }

  if (tid == 0) pairDist[p] = sRes / (float)(la + lb);
}

// ---- tiny loss reduction: 256 dists -> scalar ----
__global__ void contrastive_loss_kernel(const float* __restrict__ pd,
                                        const float* __restrict__ margin,
                                        float* __restrict__ out) {
  int w = threadIdx.x;  // one lane per work-group w<16
  float lv = 0.0f;
  if (w < NW) {
    const float* dw = pd + w * STEP;
    float aa = dw[0];
    float ssum = 0.0f;
    int nz = 1;
#pragma unroll
    for (int i = 1; i <= NG; ++i) {
      float v = dw[i] - aa;
      ssum += v;
      nz += (v != 0.0f);
    }
    float m = margin[0];
#pragma unroll
    for (int i = 1 + NG; i < STEP; ++i) {
      float v = fmaxf(m - (dw[i] - aa), 0.0f);
      ssum += v;
      nz += (v != 0.0f);
    }
    lv = ssum / (float)nz;
  }
  for (int off = 8; off >= 1; off >>= 1) lv += __shfl_down(lv, off);
  if (w == 0) out[0] = lv / (float)NW;
}

extern "C" void kernel_launch(void* const* d_in, const int* in_sizes, int n_in,
                              void* d_out, int out_size, void* d_ws,
                              size_t ws_size, hipStream_t stream) {
  (void)in_sizes; (void)n_in; (void)out_size; (void)ws_size;
  const float* data   = (const float*)d_in[0];
  const float* margin = (const float*)d_in[1];
  const int*   lens   = (const int*)d_in[2];
  float* out = (float*)d_out;
  float* pairDist = (float*)d_ws;  // 256 floats of scratch

  sdtw_pair_kernel<<<dim3(NPAIRS), dim3(NTHREADS), 0, stream>>>(data, lens,
                                                                pairDist);
  contrastive_loss_kernel<<<dim3(1), dim3(32), 0, stream>>>(pairDist, margin,
                                                            out);
}